// MAGNET_F_19353122636198
// MI455X (gfx1250) — compile-verified
//
#include <hip/hip_runtime.h>
#include <hip/hip_bf16.h>

typedef __attribute__((ext_vector_type(2))) float v2f;
typedef __attribute__((ext_vector_type(8))) float v8f;

// ---------------------------------------------------------------------------
// Problem constants (from setup_inputs): B=2, V=4, H=480, W=640, D=5, C=64
// ---------------------------------------------------------------------------
#define NIMG   10      // B + B*V images through fnet
#define H1     240
#define W1     320
#define H4     120
#define W4     160
#define CB     2       // batch
#define CV     4       // views
#define CD     5       // depth candidates
#define CC     64      // feature channels

// ---------------------------------------------------------------------------
// conv1: 3->32, 7x7, stride 2, pad 3, ReLU.  Direct conv, weights in LDS.
// Output NHWC: (10, 240, 320, 32)
// ---------------------------------------------------------------------------
__global__ void conv1_kernel(const float* __restrict__ ref,
                             const float* __restrict__ ngh,
                             const float* __restrict__ w1,
                             const float* __restrict__ b1,
                             float* __restrict__ out)
{
    __shared__ float wlds[3 * 49 * 32];   // [ (ic*49 + kh*7 + kw) * 32 + oc ]
    const int tid = threadIdx.x;
    for (int i = tid; i < 3 * 49 * 32; i += blockDim.x) {
        int oc  = i & 31;
        int idx = i >> 5;                 // ic*49 + kh*7 + kw
        int ic  = idx / 49;
        int r   = idx % 49;
        int kh  = r / 7, kw = r % 7;
        wlds[i] = w1[((oc * 3 + ic) * 7 + kh) * 7 + kw];
    }
    __syncthreads();

    const int Hin = 480, Win = 640;
    long gid = (long)blockIdx.x * blockDim.x + tid;
    const long total = (long)NIMG * H1 * W1;
    if (gid >= total) return;

    int n   = (int)(gid / (H1 * W1));
    int rem = (int)(gid % (H1 * W1));
    int oh  = rem / W1, ow = rem % W1;

    const float* img = (n < CB) ? (ref + (long)n * 3 * Hin * Win)
                                : (ngh + (long)(n - CB) * 3 * Hin * Win);

    float acc[32];
#pragma unroll
    for (int oc = 0; oc < 32; ++oc) acc[oc] = b1[oc];

    for (int ic = 0; ic < 3; ++ic) {
        const float* ch = img + (long)ic * Hin * Win;
        for (int kh = 0; kh < 7; ++kh) {
            int ih = oh * 2 - 3 + kh;
            if (ih < 0 || ih >= Hin) continue;
            for (int kw = 0; kw < 7; ++kw) {
                int iw = ow * 2 - 3 + kw;
                if (iw < 0 || iw >= Win) continue;
                float x = ch[(long)ih * Win + iw];
                const float* wr = &wlds[(ic * 49 + kh * 7 + kw) * 32];
#pragma unroll
                for (int oc = 0; oc < 32; ++oc) acc[oc] = fmaf(x, wr[oc], acc[oc]);
            }
        }
    }

    float* orow = out + gid * 32;
#pragma unroll
    for (int oc = 0; oc < 32; ++oc) orow[oc] = fmaxf(acc[oc], 0.0f);
}

// ---------------------------------------------------------------------------
// Repack OIHW (64,Cin,3,3) -> Wp[m][k], k = (kh*3+kw)*Cin + ic
// ---------------------------------------------------------------------------
__global__ void repack_w_kernel(const float* __restrict__ w,
                                float* __restrict__ Wp, int Cin)
{
    int K = Cin * 9;
    int i = blockIdx.x * blockDim.x + threadIdx.x;
    if (i >= 64 * K) return;
    int m = i / K, k = i % K;
    int s = k / Cin, ic = k % Cin;
    int kh = s / 3, kw = s % 3;
    Wp[i] = w[((m * Cin + ic) * 3 + kh) * 3 + kw];
}

// ---------------------------------------------------------------------------
// Implicit-GEMM 3x3 conv (pad 1) with fp32 WMMA 16x16x4.
// One wave computes the full 64(out-chan) x 16(pixel) tile: 4 accumulators,
// B fragment loaded once per K-step and reused by 4 wmma ops.
// in : NHWC (10, Hin, Win, CIN);  Wp: [64][CIN*9];  out: NHWC (10,Hout,Wout,64)
// ---------------------------------------------------------------------------
template <int CIN, int STRIDE, int RELU>
__global__ void conv_wmma_kernel(const float* __restrict__ in,
                                 const float* __restrict__ Wp,
                                 const float* __restrict__ bias,
                                 float* __restrict__ out,
                                 int Hin, int Win, int Hout, int Wout)
{
    const int Cout = 64;
    const int K    = CIN * 9;

    int lane = threadIdx.x & 31;
    int widb = threadIdx.x >> 5;
    long ptile = (long)blockIdx.x * (blockDim.x >> 5) + widb;

    long totalPix  = (long)NIMG * Hout * Wout;
    long npixtiles = (totalPix + 15) >> 4;
    if (ptile >= npixtiles) return;

    int half = lane >> 4;        // 0: K sub-cols {0,1}, 1: {2,3}
    int l16  = lane & 15;
    int koff = half * 2;

    // pixel owned by this lane (B-matrix column / D-matrix column)
    long p      = ptile * 16 + l16;
    bool pvalid = (p < totalPix);
    long pc     = pvalid ? p : (totalPix - 1);
    int n   = (int)(pc / ((long)Hout * Wout));
    int rem = (int)(pc % ((long)Hout * Wout));
    int oh  = rem / Wout, ow = rem % Wout;

    // A-matrix rows for this lane (one per 16-channel M-tile)
    const float* wrow0 = Wp + (long)(l16)      * K;
    const float* wrow1 = Wp + (long)(l16 + 16) * K;
    const float* wrow2 = Wp + (long)(l16 + 32) * K;
    const float* wrow3 = Wp + (long)(l16 + 48) * K;

    v8f c0 = {0.f,0.f,0.f,0.f,0.f,0.f,0.f,0.f};
    v8f c1 = c0, c2 = c0, c3 = c0;

    for (int s = 0; s < 9; ++s) {
        int kh = s / 3, kw = s % 3;
        int ih = oh * STRIDE - 1 + kh;
        int iw = ow * STRIDE - 1 + kw;
        bool inb = (ih >= 0) && (ih < Hin) && (iw >= 0) && (iw < Win);
        const float* brow = in + (((long)n * Hin + ih) * Win + iw) * CIN;
        int kbase = s * CIN;
#pragma unroll
        for (int ic = 0; ic < CIN; ic += 4) {
            int kk = kbase + ic + koff;
            v2f b;
            if (inb) b = *reinterpret_cast<const v2f*>(brow + ic + koff);
            else     { b.x = 0.f; b.y = 0.f; }
            v2f a0 = *reinterpret_cast<const v2f*>(wrow0 + kk);
            v2f a1 = *reinterpret_cast<const v2f*>(wrow1 + kk);
            v2f a2 = *reinterpret_cast<const v2f*>(wrow2 + kk);
            v2f a3 = *reinterpret_cast<const v2f*>(wrow3 + kk);
            c0 = __builtin_amdgcn_wmma_f32_16x16x4_f32(false, a0, false, b, (short)0, c0, false, false);
            c1 = __builtin_amdgcn_wmma_f32_16x16x4_f32(false, a1, false, b, (short)0, c1, false, false);
            c2 = __builtin_amdgcn_wmma_f32_16x16x4_f32(false, a2, false, b, (short)0, c2, false, false);
            c3 = __builtin_amdgcn_wmma_f32_16x16x4_f32(false, a3, false, b, (short)0, c3, false, false);
        }
    }

    if (pvalid) {
        float* orow = out + pc * Cout;
#pragma unroll
        for (int r = 0; r < 8; ++r) {
            int m = r + half * 8;           // C/D layout: VGPR r -> M = r + 8*half
            float v0 = c0[r] + bias[m];
            float v1 = c1[r] + bias[m + 16];
            float v2 = c2[r] + bias[m + 32];
            float v3 = c3[r] + bias[m + 48];
            if (RELU) {
                v0 = fmaxf(v0, 0.f); v1 = fmaxf(v1, 0.f);
                v2 = fmaxf(v2, 0.f); v3 = fmaxf(v3, 0.f);
            }
            orow[m]      = v0;
            orow[m + 16] = v1;
            orow[m + 32] = v2;
            orow[m + 48] = v3;
        }
    }
}

// ---------------------------------------------------------------------------
// Plane-sweep warp + bilinear sample + cost volume.
// feat: NHWC (10, 120, 160, 64). imgs 0..1 = ref, 2..9 = neighbors.
// out:  (2, 5, 120, 160)
// ---------------------------------------------------------------------------
__global__ void costvol_kernel(const float* __restrict__ feat,
                               const float* __restrict__ poses,     // (2,4,4,4)
                               const unsigned char* __restrict__ is_valid, // (2,4) bool
                               const float* __restrict__ Kmat,      // (2,3,3)
                               const float* __restrict__ d_center,  // (2,5,120,160)
                               float* __restrict__ outv)            // (2,5,120,160)
{
    int gid   = blockIdx.x * blockDim.x + threadIdx.x;
    int total = CB * CD * H4 * W4;
    if (gid >= total) return;

    int w = gid % W4;
    int t = gid / W4;
    int h = t % H4;  t /= H4;
    int d = t % CD;
    int b = t / CD;
    (void)d;

    // Camera intrinsics and analytic inverse (adjugate / det)
    const float* Kb = Kmat + b * 9;
    float k00 = Kb[0], k01 = Kb[1], k02 = Kb[2];
    float k10 = Kb[3], k11 = Kb[4], k12 = Kb[5];
    float k20 = Kb[6], k21 = Kb[7], k22 = Kb[8];
    float det = k00 * (k11 * k22 - k12 * k21)
              - k01 * (k10 * k22 - k12 * k20)
              + k02 * (k10 * k21 - k11 * k20);
    float id  = 1.0f / det;
    float i00 = (k11 * k22 - k12 * k21) * id, i01 = (k02 * k21 - k01 * k22) * id, i02 = (k01 * k12 - k02 * k11) * id;
    float i10 = (k12 * k20 - k10 * k22) * id, i11 = (k00 * k22 - k02 * k20) * id, i12 = (k02 * k10 - k00 * k12) * id;
    float i20 = (k10 * k21 - k11 * k20) * id, i21 = (k01 * k20 - k00 * k21) * id, i22 = (k00 * k11 - k01 * k10) * id;

    float x = (float)w, y = (float)h;
    float rx = i00 * x + i01 * y + i02;
    float ry = i10 * x + i11 * y + i12;
    float rz = i20 * x + i21 * y + i22;

    float dc = d_center[gid];            // same (B,D,H,W) layout as output
    float Px = dc * rx, Py = dc * ry, Pz = dc * rz;

    const float* refrow = feat + (((long)b * H4 + h) * W4 + w) * CC;

    float cost_sum = 0.f, mask_sum = 0.f;

    for (int v = 0; v < CV; ++v) {
        const float* M = poses + (long)(b * CV + v) * 16;  // row-major 4x4
        float Xx = M[0] * Px + M[1] * Py + M[2]  * Pz + M[3];
        float Xy = M[4] * Px + M[5] * Py + M[6]  * Pz + M[7];
        float Xz = M[8] * Px + M[9] * Py + M[10] * Pz + M[11];

        float px = k00 * Xx + k01 * Xy + k02 * Xz;
        float py = k10 * Xx + k11 * Xy + k12 * Xz;
        float pz = k20 * Xx + k21 * Xy + k22 * Xz;

        bool z_ok = pz > 0.001f;
        float zc  = fmaxf(pz, 0.001f);
        float u   = px / zc;
        float vv  = py / zc;
        bool inb  = (u >= 0.f) && (u <= (float)(W4 - 1)) &&
                    (vv >= 0.f) && (vv <= (float)(H4 - 1));
        bool valid = (is_valid[b * CV + v] != 0);
        if (!(valid && z_ok && inb)) continue;

        mask_sum += 1.f;

        float u0 = floorf(u), v0 = floorf(vv);
        float du = u - u0, dv = vv - v0;
        int iu0 = (int)u0, iv0 = (int)v0;

        const float* base = feat + (long)(CB + b * CV + v) * H4 * W4 * CC;

        int iu1 = iu0 + 1, iv1 = iv0 + 1;
        bool ok00 = (iu0 >= 0) && (iu0 < W4) && (iv0 >= 0) && (iv0 < H4);
        bool ok01 = (iu1 >= 0) && (iu1 < W4) && (iv0 >= 0) && (iv0 < H4);
        bool ok10 = (iu0 >= 0) && (iu0 < W4) && (iv1 >= 0) && (iv1 < H4);
        bool ok11 = (iu1 >= 0) && (iu1 < W4) && (iv1 >= 0) && (iv1 < H4);

        int cu0 = iu0 < 0 ? 0 : (iu0 > W4 - 1 ? W4 - 1 : iu0);
        int cu1 = iu1 < 0 ? 0 : (iu1 > W4 - 1 ? W4 - 1 : iu1);
        int cv0 = iv0 < 0 ? 0 : (iv0 > H4 - 1 ? H4 - 1 : iv0);
        int cv1 = iv1 < 0 ? 0 : (iv1 > H4 - 1 ? H4 - 1 : iv1);

        const float* q00 = base + ((long)cv0 * W4 + cu0) * CC;
        const float* q01 = base + ((long)cv0 * W4 + cu1) * CC;
        const float* q10 = base + ((long)cv1 * W4 + cu0) * CC;
        const float* q11 = base + ((long)cv1 * W4 + cu1) * CC;

        float w00 = ok00 ? (1.f - du) * (1.f - dv) : 0.f;
        float w01 = ok01 ? du * (1.f - dv) : 0.f;
        float w10 = ok10 ? (1.f - du) * dv : 0.f;
        float w11 = ok11 ? du * dv : 0.f;

        float acc = 0.f;
#pragma unroll 8
        for (int c = 0; c < CC; ++c) {
            float val = q00[c] * w00 + q01[c] * w01 + q10[c] * w10 + q11[c] * w11;
            acc = fmaf(val, refrow[c], acc);
        }
        cost_sum += acc * (1.0f / 64.0f);
    }

    outv[gid] = cost_sum / fmaxf(mask_sum, 1.0f);
}

// ---------------------------------------------------------------------------
// Launch
// ---------------------------------------------------------------------------
extern "C" void kernel_launch(void* const* d_in, const int* in_sizes, int n_in,
                              void* d_out, int out_size, void* d_ws, size_t ws_size,
                              hipStream_t stream)
{
    const float*         ref      = (const float*)d_in[0];
    const float*         ngh      = (const float*)d_in[1];
    const float*         poses    = (const float*)d_in[2];
    const unsigned char* is_valid = (const unsigned char*)d_in[3]; // bool (1B)
    const float*         Kmat     = (const float*)d_in[4];
    const float*         dcen     = (const float*)d_in[5];
    const float*         w1       = (const float*)d_in[6];
    const float*         b1       = (const float*)d_in[7];
    const float*         w2       = (const float*)d_in[8];
    const float*         b2       = (const float*)d_in[9];
    const float*         w3       = (const float*)d_in[10];
    const float*         b3       = (const float*)d_in[11];
    float* out = (float*)d_out;
    float* ws  = (float*)d_ws;

    // Workspace layout (floats)
    size_t o = 0;
    float* feat1 = ws + o; o += (size_t)NIMG * H1 * W1 * 32;  // 24.58 M
    float* feat2 = ws + o; o += (size_t)NIMG * H4 * W4 * 64;  // 12.29 M
    float* feat3 = ws + o; o += (size_t)NIMG * H4 * W4 * 64;  // 12.29 M
    float* Wp2   = ws + o; o += (size_t)64 * 32 * 9;
    float* Wp3   = ws + o; o += (size_t)64 * 64 * 9;
    (void)ws_size; (void)in_sizes; (void)n_in; (void)out_size;

    // conv1 (direct, fp32)
    {
        long total  = (long)NIMG * H1 * W1;
        int  blocks = (int)((total + 255) / 256);
        conv1_kernel<<<blocks, 256, 0, stream>>>(ref, ngh, w1, b1, feat1);
    }

    // weight repack for implicit GEMM
    repack_w_kernel<<<(64 * 288 + 255) / 256, 256, 0, stream>>>(w2, Wp2, 32);
    repack_w_kernel<<<(64 * 576 + 255) / 256, 256, 0, stream>>>(w3, Wp3, 64);

    // conv2/conv3 via fp32 WMMA implicit GEMM (one wave = 64ch x 16px tile)
    {
        long totalPix = (long)NIMG * H4 * W4;
        long waves    = (totalPix + 15) / 16;      // one wave per 16-pixel tile
        long threads  = waves * 32;
        int  blocks   = (int)((threads + 255) / 256);
        // conv2 (32->64, 3x3 s2 p1, ReLU)
        conv_wmma_kernel<32, 2, 1><<<blocks, 256, 0, stream>>>(
            feat1, Wp2, b2, feat2, H1, W1, H4, W4);
        // conv3 (64->64, 3x3 s1 p1, no activation)
        conv_wmma_kernel<64, 1, 0><<<blocks, 256, 0, stream>>>(
            feat2, Wp3, b3, feat3, H4, W4, H4, W4);
    }

    // warp + bilinear + cost volume
    {
        int total  = CB * CD * H4 * W4;
        int blocks = (total + 255) / 256;
        costvol_kernel<<<blocks, 256, 0, stream>>>(feat3, poses, is_valid,
                                                   Kmat, dcen, out);
    }
}